// SRUCell_163208757302
// MI455X (gfx1250) — compile-verified
//
#include <hip/hip_runtime.h>
#include <hip/hip_bf16.h>

// SRU on gfx1250 (MI455X):
//   Phase 1 (per time-chunk): U = X_chunk @ W via V_WMMA_F32_16X16X4_F32
//   Phase 2 (per time-chunk): sequential scan, parallel over B*D channels.
//
// L=2048, B=32, D=512, weight (512, 1536), u reshaped (L,B,D,3) so
// u[t,b,d,j] = U_row(t*B+b)[3*d + j].

#define SRU_L 2048
#define SRU_B 32
#define SRU_D 512
#define SRU_N 1536   // 3*D

typedef __attribute__((ext_vector_type(2))) float v2f;
typedef __attribute__((ext_vector_type(8))) float v8f;

// ---------------------------------------------------------------------------
// GEMM: U[Mrows x 1536] = X[Mrows x 512] * W[512 x 1536], f32 WMMA.
// One wave computes a 16(M) x 64(N) tile (4 accumulators) with a K-loop of
// 128 steps of 16x16x4.
// A layout (16x4 f32): lanes 0-15 -> M=lane, VGPR0=K+0, VGPR1=K+1;
//                      lanes 16-31 -> M=lane-16, VGPR0=K+2, VGPR1=K+3.
// B layout (4x16 f32, row-striped like C/D): VGPR0 -> rows K+0 (lanes 0-15)
//                      and K+2 (lanes 16-31); VGPR1 -> rows K+1 / K+3.
// C/D layout (16x16 f32): VGPR v -> row v (lanes 0-15) / row v+8 (lanes 16-31),
//                         col = lane & 15.
// ---------------------------------------------------------------------------
__global__ __launch_bounds__(256)
void sru_gemm_wmma_f32(const float* __restrict__ X,   // (Mrows, 512)
                       const float* __restrict__ W,   // (512, 1536)
                       float* __restrict__ U,         // (Mrows, 1536)
                       int Mrows) {
  const int lane  = threadIdx.x & 31;
  const int wave  = blockIdx.x * (blockDim.x >> 5) + (threadIdx.x >> 5);
  const int NG    = SRU_N / 64;            // 24 N-groups of 64 columns
  const int mtile = wave / NG;
  const int ng    = wave % NG;
  const int m0    = mtile * 16;
  if (m0 >= Mrows) return;
  const int n0    = ng * 64;

  const int half  = lane >> 4;             // 0: lanes 0-15, 1: lanes 16-31
  const int l15   = lane & 15;

  // Per-lane A row pointer; this lane reads K offsets {2*half, 2*half+1}.
  const float* __restrict__ Arow = X + (size_t)(m0 + l15) * SRU_D + 2 * half;

  v8f acc0 = {}, acc1 = {}, acc2 = {}, acc3 = {};

  for (int k = 0; k < SRU_D; k += 4) {
    // A fragment: 2 contiguous f32 -> one global_load_b64 per lane.
    v2f a = *(const v2f*)(Arow + k);

    // B fragments for the 4 N-subtiles.
    const float* __restrict__ Wr0 = W + (size_t)(k + 2 * half) * SRU_N + n0 + l15;
    const float* __restrict__ Wr1 = Wr0 + SRU_N;
    v2f b0 = { Wr0[0],  Wr1[0]  };
    v2f b1 = { Wr0[16], Wr1[16] };
    v2f b2 = { Wr0[32], Wr1[32] };
    v2f b3 = { Wr0[48], Wr1[48] };

    acc0 = __builtin_amdgcn_wmma_f32_16x16x4_f32(false, a, false, b0, (short)0, acc0, false, false);
    acc1 = __builtin_amdgcn_wmma_f32_16x16x4_f32(false, a, false, b1, (short)0, acc1, false, false);
    acc2 = __builtin_amdgcn_wmma_f32_16x16x4_f32(false, a, false, b2, (short)0, acc2, false, false);
    acc3 = __builtin_amdgcn_wmma_f32_16x16x4_f32(false, a, false, b3, (short)0, acc3, false, false);
  }

  // Store row-major: VGPR v -> row m0 + v + 8*half, col n0 + 16*tile + l15.
  float* __restrict__ Ub = U + (size_t)(m0 + 8 * half) * SRU_N + n0 + l15;
#pragma unroll
  for (int v = 0; v < 8; ++v) {
    float* __restrict__ row = Ub + (size_t)v * SRU_N;
    row[0]  = acc0[v];
    row[16] = acc1[v];
    row[32] = acc2[v];
    row[48] = acc3[v];
  }
}

// ---------------------------------------------------------------------------
// Scan: one thread per (b, d) channel; sequential over the chunk's steps.
// u[t,b,d,j] = U[(t*B+b)*1536 + 3*d + j]  (12 contiguous bytes per lane).
// ---------------------------------------------------------------------------
__device__ __forceinline__ float sru_sigmoid(float x) {
  return 1.0f / (1.0f + __expf(-x));
}
__device__ __forceinline__ float sru_tanh(float x) {
  // tanh(x) = 1 - 2/(exp(2x)+1), via the hardware v_exp_f32 path.
  return 1.0f - 2.0f / (__expf(2.0f * x) + 1.0f);
}

__global__ __launch_bounds__(256)
void sru_scan(const float* __restrict__ U,        // (Lcur*B, 1536)
              const float* __restrict__ X,        // inputs + l0*B*D
              const float* __restrict__ init_c,   // (B*D)
              const float* __restrict__ bias,     // (2*D)
              float* __restrict__ h_out,          // d_out + l0*B*D
              float* __restrict__ c_state,        // ws carry (B*D)
              float* __restrict__ c_last_out,     // d_out tail on final chunk, else null
              int Lcur, int first_chunk) {
  const int tid = blockIdx.x * blockDim.x + threadIdx.x;
  if (tid >= SRU_B * SRU_D) return;
  const int b = tid / SRU_D;
  const int d = tid % SRU_D;

  const float bf = bias[d];
  const float br = bias[SRU_D + d];
  float c = first_chunk ? init_c[tid] : c_state[tid];

  const float* __restrict__ Up = U + (size_t)b * SRU_N + 3 * d;
  const float* __restrict__ Xp = X + (size_t)b * SRU_D + d;
  float* __restrict__ Hp       = h_out + (size_t)b * SRU_D + d;

  for (int t = 0; t < Lcur; ++t) {
    const float u0 = Up[0];
    const float u1 = Up[1];
    const float u2 = Up[2];
    const float x  = *Xp;

    const float f = sru_sigmoid(u1 + bf);
    const float r = sru_sigmoid(u2 + br);
    c = f * c + (1.0f - f) * u0;
    *Hp = r * sru_tanh(c) + (1.0f - r) * x;

    Up += (size_t)SRU_B * SRU_N;
    Xp += SRU_B * SRU_D;
    Hp += SRU_B * SRU_D;
  }

  c_state[tid] = c;
  if (c_last_out) c_last_out[tid] = c;
}

// ---------------------------------------------------------------------------
// Launch: chunk the L dimension so the U scratch fits in ws_size.
// ws layout: [c_state: B*D f32][pad to 256B][U chunk: Lc*B*1536 f32]
// ---------------------------------------------------------------------------
extern "C" void kernel_launch(void* const* d_in, const int* in_sizes, int n_in,
                              void* d_out, int out_size, void* d_ws, size_t ws_size,
                              hipStream_t stream) {
  const float* inputs = (const float*)d_in[0];   // (L, B, D)
  const float* init_c = (const float*)d_in[1];   // (B, D)
  const float* weight = (const float*)d_in[2];   // (D, 3D)
  const float* bias   = (const float*)d_in[3];   // (2D)

  float* out    = (float*)d_out;
  float* h_out  = out;                                   // (L, B, D)
  float* c_last = out + (size_t)SRU_L * SRU_B * SRU_D;   // (B, D)

  const size_t state_bytes = (size_t)SRU_B * SRU_D * sizeof(float);      // 64 KB
  const size_t state_pad   = (state_bytes + 255) & ~(size_t)255;
  float* c_state = (float*)d_ws;
  float* Ubuf    = (float*)((char*)d_ws + state_pad);

  const size_t per_step = (size_t)SRU_B * SRU_N * sizeof(float);         // 192 KB/step
  long long avail = (long long)ws_size - (long long)state_pad;
  int Lc = (avail > 0) ? (int)(avail / (long long)per_step) : 0;
  if (Lc < 1) Lc = 1;
  if (Lc > SRU_L) Lc = SRU_L;

  for (int l0 = 0; l0 < SRU_L; l0 += Lc) {
    const int Lcur  = (SRU_L - l0 < Lc) ? (SRU_L - l0) : Lc;
    const int Mrows = Lcur * SRU_B;                 // multiple of 32
    const int mtiles = Mrows / 16;
    const int waves  = mtiles * (SRU_N / 64);       // 48 * Lcur
    const int blocks = (waves + 7) / 8;             // 8 waves (256 thr) per block

    sru_gemm_wmma_f32<<<blocks, 256, 0, stream>>>(
        inputs + (size_t)l0 * SRU_B * SRU_D, weight, Ubuf, Mrows);

    const int first_chunk = (l0 == 0) ? 1 : 0;
    const int last_chunk  = (l0 + Lcur == SRU_L);
    sru_scan<<<(SRU_B * SRU_D) / 256, 256, 0, stream>>>(
        Ubuf,
        inputs + (size_t)l0 * SRU_B * SRU_D,
        init_c, bias,
        h_out + (size_t)l0 * SRU_B * SRU_D,
        c_state,
        last_chunk ? c_last : nullptr,
        Lcur, first_chunk);
  }
}